// Model_21766894256248
// MI455X (gfx1250) — compile-verified
//
#include <hip/hip_runtime.h>
#include <stdint.h>

// SHA-256 (single 64-byte block, no padding append — matches reference) over
// B=262144 messages. Input: B*64 int64 elements, one byte value each.
// Output: B*8 int64 elements (32-bit hash words zero-extended).
//
// gfx1250 data path: Tensor Data Mover. Each block of 128 threads owns 128
// messages (64 KB of int64-inflated input). ONE tensor_load_to_lds per block
// gathers the little-endian low byte of each int64 (1-byte elements, 8-byte
// stride) into a dense 8 KB LDS image, tracked on TENSORcnt. This replaces
// 256 per-wave async issues + ~130 VALU/thread of address bookkeeping.
// Fallback (builtin unavailable): per-lane global_load_async_to_lds_b8.

#define THREADS 128

typedef unsigned int u32x4 __attribute__((ext_vector_type(4)));
typedef int          i32x4 __attribute__((ext_vector_type(4)));
typedef int          i32x8 __attribute__((ext_vector_type(8)));

__device__ __forceinline__ uint32_t rotr32(uint32_t x, int n) {
    return (x >> n) | (x << (32 - n));   // -> v_alignbit_b32
}

__global__ __launch_bounds__(THREADS) void sha256_batch(
        const unsigned long long* __restrict__ msg,  // B*64, one byte per int64
        unsigned long long* __restrict__ out,        // B*8 hash words
        long long nmsg) {
    static constexpr uint32_t K[64] = {
        0x428a2f98u,0x71374491u,0xb5c0fbcfu,0xe9b5dba5u,0x3956c25bu,0x59f111f1u,0x923f82a4u,0xab1c5ed5u,
        0xd807aa98u,0x12835b01u,0x243185beu,0x550c7dc3u,0x72be5d74u,0x80deb1feu,0x9bdc06a7u,0xc19bf174u,
        0xe49b69c1u,0xefbe4786u,0x0fc19dc6u,0x240ca1ccu,0x2de92c6fu,0x4a7484aau,0x5cb0a9dcu,0x76f988dau,
        0x983e5152u,0xa831c66du,0xb00327c8u,0xbf597fc7u,0xc6e00bf3u,0xd5a79147u,0x06ca6351u,0x14292967u,
        0x27b70a85u,0x2e1b2138u,0x4d2c6dfcu,0x53380d13u,0x650a7354u,0x766a0abbu,0x81c2c92eu,0x92722c85u,
        0xa2bfe8a1u,0xa81a664bu,0xc24b8b70u,0xc76c51a3u,0xd192e819u,0xd6990624u,0xf40e3585u,0x106aa070u,
        0x19a4c116u,0x1e376c08u,0x2748774cu,0x34b0bcb5u,0x391c0cb3u,0x4ed8aa4au,0x5b9cca4fu,0x682e6ff3u,
        0x748f82eeu,0x78a5636fu,0x84c87814u,0x8cc70208u,0x90befffau,0xa4506cebu,0xbef9a3f7u,0xc67178f2u};

    __shared__ uint32_t lds[THREADS * 16];            // dense: 64 B per message

    const int t = threadIdx.x;
    const long long mBlock = (long long)blockIdx.x * THREADS;
    const long long gm = mBlock + t;                  // this thread's message
    const bool fullBlock = (mBlock + THREADS) <= nmsg; // block-uniform

    uint32_t W[16];

    if (fullBlock) {
        // Low 32 bits of a flat LDS pointer == LDS byte offset.
        const uint32_t ldsBase = (uint32_t)(uintptr_t)(void*)&lds[0];
        const unsigned long long ga =
            (unsigned long long)(uintptr_t)(msg + mBlock * 64);

#if defined(__has_builtin) && __has_builtin(__builtin_amdgcn_tensor_load_to_lds)
        if (t < 32) {   // one wave issues the DMA (EXEC ignored by TDM)
            // ---- Tensor DMA descriptor (ISA 8.3-8.6), all block-uniform ----
            // 3-D tile: x=1 elem (1B), y=64 (stride 8B), z=128 (stride 512B).
            const u32x4 g0 = {
                1u,                                        // count=1, user D#
                ldsBase,                                   // lds_addr
                (uint32_t)ga,                              // global_addr[31:0]
                ((uint32_t)(ga >> 32) & 0x01FFFFFFu) | (2u << 30) }; // +type=2
            const i32x8 g1 = {
                0,                                         // mask=0, 1B data, no pad
                (int)(1u << 16),                           // tensor_dim0 = 1
                (int)(64u << 16),                          // tensor_dim1 = 64
                (int)(1u << 16),                           // tile_dim0 = 1
                (int)(64u | (128u << 16)),                 // tile_dim1=64, tile_dim2=128
                8,                                         // tensor_dim0_stride = 8
                (int)(512u << 16),                         // tensor_dim1_stride = 512
                0 };
            const i32x4 g2 = { 128, 0, 0, 0 };             // tensor_dim2 = 128
            const i32x4 g3 = { 0, 0, 0, 0 };
#if __clang_major__ >= 23
            const i32x8 gz = { 0, 0, 0, 0, 0, 0, 0, 0 };
            __builtin_amdgcn_tensor_load_to_lds(g0, g1, g2, g3, gz, 0);
#else
            __builtin_amdgcn_tensor_load_to_lds(g0, g1, g2, g3, 0);
#endif
            __builtin_amdgcn_s_wait_tensorcnt((short)0);   // DMA complete
        }
#else
        // ---- fallback: per-lane async byte gather (dense LDS layout) ----
        uint32_t voff = (uint32_t)t * 8u;
        uint32_t dst  = ldsBase + (uint32_t)t;
        #pragma unroll
        for (int j = 0; j < 64; ++j) {
            asm volatile("global_load_async_to_lds_b8 %0, %1, %2"
                         :: "v"(dst), "v"(voff), "s"(ga) : "memory");
            voff += (uint32_t)(THREADS * 8);
            dst  += (uint32_t)THREADS;
        }
        asm volatile("s_wait_asynccnt 0x0" ::: "memory");
#endif
        __syncthreads();                                   // LDS visible to all

        // ---- 16 big-endian words from dense LDS ----
        const int base = t * 16;
        #pragma unroll
        for (int w = 0; w < 16; ++w)
            W[w] = __builtin_bswap32(lds[base + w]);       // v_perm_b32
    } else {
        // Rare tail block (never taken for B=262144): direct global gather.
        __syncthreads();
        if (gm < nmsg) {
            const unsigned long long* p = msg + gm * 64;
            #pragma unroll
            for (int w = 0; w < 16; ++w) {
                const uint32_t b0 = (uint32_t)p[4 * w + 0];
                const uint32_t b1 = (uint32_t)p[4 * w + 1];
                const uint32_t b2 = (uint32_t)p[4 * w + 2];
                const uint32_t b3 = (uint32_t)p[4 * w + 3];
                W[w] = (b0 << 24) | (b1 << 16) | (b2 << 8) | b3;
            }
        }
    }

    if (gm < nmsg) {
        // ---------- 64 rounds, fully unrolled, rolling schedule ----------
        uint32_t a = 0x6a09e667u, b2 = 0xbb67ae85u, c = 0x3c6ef372u, d = 0xa54ff53au;
        uint32_t e = 0x510e527fu, f  = 0x9b05688cu, g = 0x1f83d9abu, h = 0x5be0cd19u;
        #pragma unroll
        for (int i = 0; i < 64; ++i) {
            uint32_t wi;
            if (i < 16) {
                wi = W[i];
            } else {
                const uint32_t x15 = W[(i - 15) & 15];
                const uint32_t x2  = W[(i - 2) & 15];
                const uint32_t s0 = rotr32(x15, 7) ^ rotr32(x15, 18) ^ (x15 >> 3);
                const uint32_t s1 = rotr32(x2, 17) ^ rotr32(x2, 19) ^ (x2 >> 10);
                wi = W[i & 15] + s0 + W[(i - 7) & 15] + s1;
                W[i & 15] = wi;
            }
            const uint32_t S1  = rotr32(e, 6) ^ rotr32(e, 11) ^ rotr32(e, 25);
            const uint32_t ch  = (e & f) ^ (~e & g);           // -> v_bitop3
            const uint32_t t1  = h + S1 + ch + K[i] + wi;
            const uint32_t S0  = rotr32(a, 2) ^ rotr32(a, 13) ^ rotr32(a, 22);
            const uint32_t maj = (a & b2) ^ (a & c) ^ (b2 & c); // -> v_bitop3
            const uint32_t t2  = S0 + maj;
            h = g; g = f; f = e; e = d + t1;
            d = c; c = b2; b2 = a; a = t1 + t2;
        }

        unsigned long long* o = out + gm * 8;
        o[0] = (unsigned long long)(uint32_t)(0x6a09e667u + a);
        o[1] = (unsigned long long)(uint32_t)(0xbb67ae85u + b2);
        o[2] = (unsigned long long)(uint32_t)(0x3c6ef372u + c);
        o[3] = (unsigned long long)(uint32_t)(0xa54ff53au + d);
        o[4] = (unsigned long long)(uint32_t)(0x510e527fu + e);
        o[5] = (unsigned long long)(uint32_t)(0x9b05688cu + f);
        o[6] = (unsigned long long)(uint32_t)(0x1f83d9abu + g);
        o[7] = (unsigned long long)(uint32_t)(0x5be0cd19u + h);
    }
}

extern "C" void kernel_launch(void* const* d_in, const int* in_sizes, int n_in,
                              void* d_out, int out_size, void* d_ws, size_t ws_size,
                              hipStream_t stream) {
    (void)n_in; (void)out_size; (void)d_ws; (void)ws_size;
    const unsigned long long* msg = (const unsigned long long*)d_in[0];
    unsigned long long* out = (unsigned long long*)d_out;
    const long long nmsg = (long long)in_sizes[0] / 64;   // B = 262144
    const int blocks = (int)((nmsg + THREADS - 1) / THREADS);
    sha256_batch<<<blocks, THREADS, 0, stream>>>(msg, out, nmsg);
}